// KNNGraphBuilder_23167053595055
// MI455X (gfx1250) — compile-verified
//
#include <hip/hip_runtime.h>
#include <hip/hip_bf16.h>
#include <stdint.h>

// ---------------------------------------------------------------------------
// Problem constants (match reference: x = (8192, 512) f32, top-K = 16)
// ---------------------------------------------------------------------------
#define N_PTS   8192
#define DIMS    512
#define K_TOP   16

// GEMM tiling: block tile 128x64, 8 waves, each wave a 32x32 sub-tile (2x2 WMMA)
#define BM      128
#define BN      64
#define KB      32          // K-block staged in LDS
#define NKB     (DIMS / KB) // 16

typedef float v2f __attribute__((ext_vector_type(2)));
typedef float v4f __attribute__((ext_vector_type(4)));
typedef float v8f __attribute__((ext_vector_type(8)));

// ---------------------------------------------------------------------------
// CDNA5 async global->LDS copy (ASYNCcnt path) + waits
// ---------------------------------------------------------------------------
__device__ __forceinline__ void async_b128(uint32_t lds_off, uint64_t gaddr) {
    // VDST = per-lane LDS byte offset, VADDR = 64-bit global address
    asm volatile("global_load_async_to_lds_b128 %0, %1, off"
                 :: "v"(lds_off), "v"(gaddr) : "memory");
}
#define WAIT_ASYNC_LE6() asm volatile("s_wait_asynccnt 0x6" ::: "memory")
#define WAIT_ASYNC_0()   asm volatile("s_wait_asynccnt 0x0" ::: "memory")

__device__ __forceinline__ uint32_t lds_off_of(const void* p) {
    // generic (flat) LDS pointer: low 32 bits are the LDS byte offset
    return (uint32_t)(uintptr_t)p;
}

__device__ __forceinline__ v8f wmma_f32(v2f a, v2f b, v8f c) {
    return __builtin_amdgcn_wmma_f32_16x16x4_f32(false, a, false, b,
                                                 (short)0, c, false, false);
}

// ---------------------------------------------------------------------------
// Kernel 0: copy x into the first output region (16 MB)
// ---------------------------------------------------------------------------
__global__ void copy_x_kernel(const v4f* __restrict__ src,
                              v4f* __restrict__ dst, int n4) {
    int i = blockIdx.x * blockDim.x + threadIdx.x;
    int stride = gridDim.x * blockDim.x;
    for (; i < n4; i += stride) {
        v4f v = __builtin_nontemporal_load(&src[i]);
        __builtin_nontemporal_store(v, &dst[i]);
    }
}

// ---------------------------------------------------------------------------
// Kernel 1: sim = x * x^T  (fp32 WMMA 16x16x4), full matrix written to `sim`
// grid = (N/BN, N/BM), block = 256 threads (8 waves, 4x2 wave grid).
// Each wave computes a 32x32 sub-tile: 2x2 accumulators, 4 WMMA per 4 frags.
// ---------------------------------------------------------------------------
__global__ void __launch_bounds__(256)
gemm_sim_kernel(const float* __restrict__ x, float* __restrict__ sim) {
    __shared__ float Asm[2][BM][KB];   // rows m0..m0+127, k-block (32 KB)
    __shared__ float Bsm[2][BN][KB];   // rows n0..n0+63,  k-block (16 KB)

    const int t    = threadIdx.x;
    const int lane = t & 31;
    const int wv   = t >> 5;             // 0..7
    const int l16  = lane & 15;
    const int half = lane >> 4;          // 0 or 1
    const int mw   = (wv & 3) << 5;      // wave M offset: 0,32,64,96
    const int nw   = (wv >> 2) << 5;     // wave N offset: 0,32

    const int m0 = blockIdx.y * BM;
    const int n0 = blockIdx.x * BN;

    const uint32_t ldsA0 = lds_off_of(&Asm[0][0][0]);
    const uint32_t ldsA1 = lds_off_of(&Asm[1][0][0]);
    const uint32_t ldsB0 = lds_off_of(&Bsm[0][0][0]);
    const uint32_t ldsB1 = lds_off_of(&Bsm[1][0][0]);

    // Per thread per k-block: 4 x b128 for A (128x32 floats), 2 x b128 for B.
    auto issue = [&](int buf, int kb) {
        const int k0 = kb * KB;
        const uint32_t la = buf ? ldsA1 : ldsA0;
        const uint32_t lb = buf ? ldsB1 : ldsB0;
        #pragma unroll
        for (int p = 0; p < 4; ++p) {
            const int fa = t * 4 + p;          // 0..1023
            const int r = fa >> 3, q = fa & 7; // row 0..127, quad 0..7
            async_b128(la + (uint32_t)((r * KB + q * 4) * 4),
                       (uint64_t)(uintptr_t)(x + (size_t)(m0 + r) * DIMS + k0 + q * 4));
        }
        #pragma unroll
        for (int p = 0; p < 2; ++p) {
            const int fb = t * 2 + p;          // 0..511
            const int r = fb >> 3, q = fb & 7; // row 0..63, quad 0..7
            async_b128(lb + (uint32_t)((r * KB + q * 4) * 4),
                       (uint64_t)(uintptr_t)(x + (size_t)(n0 + r) * DIMS + k0 + q * 4));
        }
    };

    v8f c00 = {}, c01 = {}, c10 = {}, c11 = {};

    issue(0, 0);   // prologue: fill buffer 0

    for (int kb = 0; kb < NKB; ++kb) {
        const int cur = kb & 1;
        if (kb + 1 < NKB) {
            issue(cur ^ 1, kb + 1);   // prefetch next k-block (6 async ops)
            WAIT_ASYNC_LE6();         // oldest 6 (current buffer) complete
        } else {
            WAIT_ASYNC_0();
        }
        __syncthreads();              // all waves' current buffer visible

        const float* A = &Asm[cur][0][0];
        const float* B = &Bsm[cur][0][0];
        #pragma unroll
        for (int k4 = 0; k4 < KB; k4 += 4) {
            const int kk = k4 + half * 2;   // half-wave selects K pair
            v2f a0 = *(const v2f*)(A + (mw + l16)      * KB + kk);
            v2f a1 = *(const v2f*)(A + (mw + 16 + l16) * KB + kk);
            v2f b0 = *(const v2f*)(B + (nw + l16)      * KB + kk);
            v2f b1 = *(const v2f*)(B + (nw + 16 + l16) * KB + kk);
            c00 = wmma_f32(a0, b0, c00);
            c01 = wmma_f32(a0, b1, c01);
            c10 = wmma_f32(a1, b0, c10);
            c11 = wmma_f32(a1, b1, c11);
        }
        __syncthreads();              // buffer may be overwritten next iter
    }

    // C layout: VGPR j -> M = j (lanes 0-15) / M = j+8 (lanes 16-31), N = lane%16
    const size_t rowBase = (size_t)(m0 + mw + half * 8);
    const size_t col0 = (size_t)(n0 + nw + l16);
    #pragma unroll
    for (int j = 0; j < 8; ++j) {
        const size_t r0 = (rowBase + j) * (size_t)N_PTS;
        const size_t r1 = (rowBase + 16 + j) * (size_t)N_PTS;
        __builtin_nontemporal_store(c00[j], &sim[r0 + col0]);
        __builtin_nontemporal_store(c01[j], &sim[r0 + col0 + 16]);
        __builtin_nontemporal_store(c10[j], &sim[r1 + col0]);
        __builtin_nontemporal_store(c11[j], &sim[r1 + col0 + 16]);
    }
}

// ---------------------------------------------------------------------------
// Kernel 2: per-row top-16 threshold + in-place filter.
// One 256-thread block per row; thread owns 32 strided elements in registers.
// 16 iterated max-extractions (LDS tree reduce) -> 16th largest = threshold.
// ---------------------------------------------------------------------------
__global__ void __launch_bounds__(256)
topk_filter_kernel(float* __restrict__ adj) {
    const int row = blockIdx.x;
    const int t   = threadIdx.x;
    float* rp = adj + (size_t)row * N_PTS;

    float vals[32], work[32];
    #pragma unroll
    for (int i = 0; i < 32; ++i) {
        vals[i] = rp[i * 256 + t];
        work[i] = vals[i];
    }

    __shared__ float redv[256];
    __shared__ int   redi[256];

    float thr = -__builtin_inff();
    for (int iter = 0; iter < K_TOP; ++iter) {
        float bv = -__builtin_inff();
        int   bi = 0x7fffffff;
        #pragma unroll
        for (int i = 0; i < 32; ++i) {
            if (work[i] > bv) { bv = work[i]; bi = i * 256 + t; }
        }
        redv[t] = bv; redi[t] = bi;
        __syncthreads();
        #pragma unroll
        for (int s = 128; s > 0; s >>= 1) {
            if (t < s) {
                float ov = redv[t + s]; int oi = redi[t + s];
                if (ov > redv[t] || (ov == redv[t] && oi < redi[t])) {
                    redv[t] = ov; redi[t] = oi;
                }
            }
            __syncthreads();
        }
        const float wvv = redv[0];
        const int   wii = redi[0];
        thr = wvv;
        if ((wii & 255) == t) work[wii >> 8] = -__builtin_inff();
        __syncthreads();
    }

    #pragma unroll
    for (int i = 0; i < 32; ++i) {
        float v = vals[i];
        __builtin_nontemporal_store((v >= thr) ? v : 0.0f, &rp[i * 256 + t]);
    }
}

// ---------------------------------------------------------------------------
// Launcher
// ---------------------------------------------------------------------------
extern "C" void kernel_launch(void* const* d_in, const int* in_sizes, int n_in,
                              void* d_out, int out_size, void* d_ws, size_t ws_size,
                              hipStream_t stream) {
    const float* x = (const float*)d_in[0];
    float* out_x = (float*)d_out;
    float* adj   = out_x + (size_t)N_PTS * DIMS;

    // 1) copy x -> output region 0
    {
        const int n4 = N_PTS * DIMS / 4;
        copy_x_kernel<<<1024, 256, 0, stream>>>((const v4f*)x, (v4f*)out_x, n4);
    }
    // 2) full similarity matrix into adj region (scratch, fully overwritten)
    {
        dim3 grid(N_PTS / BN, N_PTS / BM);   // (128, 64)
        gemm_sim_kernel<<<grid, 256, 0, stream>>>(x, adj);
    }
    // 3) per-row top-16 filter in place
    {
        topk_filter_kernel<<<N_PTS, 256, 0, stream>>>(adj);
    }
}